// FeaturePropagation_32865089749525
// MI455X (gfx1250) — compile-verified
//
#include <hip/hip_runtime.h>
#include <hip/hip_bf16.h>

// ---------------- problem constants (from setup_inputs) ----------------
#define B_   8
#define N1_  8192
#define N2_  2048
#define C1_  64
#define C2_  256
#define CX_  320          // C1 + C2
#define O0_  256
#define O1_  128
#define EPS_BN   1e-5f
#define EPS_DIST 1e-10f

typedef __attribute__((ext_vector_type(2))) float v2f;
typedef __attribute__((ext_vector_type(8))) float v8f;

// ---------------------------------------------------------------------
// Kernel 1: three-nearest-neighbor + inverse-distance weights.
// One block = 256 query points of one batch; xyz2[b] (24 KB) lives in LDS.
// ---------------------------------------------------------------------
__global__ __launch_bounds__(256) void knn3_kernel(
    const float* __restrict__ xyz1, const float* __restrict__ xyz2,
    int* __restrict__ idx_out, float* __restrict__ w_out) {
  __shared__ float s_xyz[N2_ * 3];
  const int b = blockIdx.y;
  const int n = blockIdx.x * 256 + threadIdx.x;

  const float* x2 = xyz2 + (size_t)b * N2_ * 3;
  for (int i = threadIdx.x; i < N2_ * 3; i += 256) s_xyz[i] = x2[i];
  __syncthreads();

  const float* q = xyz1 + ((size_t)b * N1_ + n) * 3;
  const float qx = q[0], qy = q[1], qz = q[2];

  float d0 = 1e30f, d1 = 1e30f, d2 = 1e30f;
  int   i0 = 0,     i1 = 0,     i2 = 0;
  for (int m = 0; m < N2_; ++m) {
    const float dx = qx - s_xyz[3 * m + 0];
    const float dy = qy - s_xyz[3 * m + 1];
    const float dz = qz - s_xyz[3 * m + 2];
    const float d  = dx * dx + dy * dy + dz * dz;
    if (d < d0)      { d2 = d1; i2 = i1; d1 = d0; i1 = i0; d0 = d; i0 = m; }
    else if (d < d1) { d2 = d1; i2 = i1; d1 = d;  i1 = m; }
    else if (d < d2) { d2 = d;  i2 = m; }
  }
  d0 = fmaxf(d0, EPS_DIST);
  d1 = fmaxf(d1, EPS_DIST);
  d2 = fmaxf(d2, EPS_DIST);
  float w0 = 1.0f / d0, w1 = 1.0f / d1, w2 = 1.0f / d2;
  const float inv = 1.0f / (w0 + w1 + w2);
  w0 *= inv; w1 *= inv; w2 *= inv;

  const size_t o = ((size_t)b * N1_ + n) * 3;
  idx_out[o + 0] = i0; idx_out[o + 1] = i1; idx_out[o + 2] = i2;
  w_out[o + 0] = w0;   w_out[o + 1] = w1;   w_out[o + 2] = w2;
}

// ---------------------------------------------------------------------
// Kernel 2: build X = concat(points1, three_interpolate(points2)).
// One block per (channel, batch). Interp channels cache the points2 row
// (8 KB) in LDS and gather via idx.
// ---------------------------------------------------------------------
__global__ __launch_bounds__(256) void buildx_kernel(
    const float* __restrict__ points1, const float* __restrict__ points2,
    const int* __restrict__ idx, const float* __restrict__ w,
    float* __restrict__ X) {
  __shared__ float srow[N2_];
  const int c = blockIdx.x;
  const int b = blockIdx.y;
  float* xrow = X + ((size_t)b * CX_ + c) * N1_;

  if (c < C1_) {
    const float* src = points1 + ((size_t)b * C1_ + c) * N1_;
    for (int n = threadIdx.x; n < N1_; n += 256) xrow[n] = src[n];
  } else {
    const float* prow = points2 + ((size_t)b * C2_ + (c - C1_)) * N2_;
    for (int i = threadIdx.x; i < N2_; i += 256) srow[i] = prow[i];
    __syncthreads();
    const int*   ib = idx + (size_t)b * N1_ * 3;
    const float* wb = w   + (size_t)b * N1_ * 3;
    for (int n = threadIdx.x; n < N1_; n += 256) {
      const int j0 = ib[3 * n], j1 = ib[3 * n + 1], j2 = ib[3 * n + 2];
      xrow[n] = wb[3 * n]     * srow[j0]
              + wb[3 * n + 1] * srow[j1]
              + wb[3 * n + 2] * srow[j2];
    }
  }
}

// ---------------------------------------------------------------------
// Kernel 3: Y[b] = Wm(OxK) * X[b](KxN) + bias via V_WMMA_F32_16X16X4_F32.
// 256 threads = 8 waves; each wave owns a 16(M) x 128(N) tile (8 accums).
// X K-chunks (16 x 128 floats) are double-buffered in LDS in a
// pair-interleaved layout so each B fragment is a single ds_load_b64:
//   element (k, col) lives at sB[buf][((k>>1)*128 + col)*2 + (k&1)].
// A fragments for the whole chunk are hoisted (4 global_load_b64 issued
// together). One barrier per chunk. EXEC uniform as WMMA requires.
// Requires K % 16 == 0, N % 128 == 0, O % 128 == 0 (per grid.y tiling).
// ---------------------------------------------------------------------
__global__ __launch_bounds__(256) void gemm_bias_kernel(
    const float* __restrict__ Wm, const float* __restrict__ bias,
    const float* __restrict__ Xin, float* __restrict__ Yout,
    int O, int K, int N) {
  __shared__ float sB[2][16 * 128];
  const int tid  = threadIdx.x;
  const int wave = tid >> 5;
  const int lane = tid & 31;
  const int half = lane >> 4;   // 0: lanes 0-15, 1: lanes 16-31
  const int lid  = lane & 15;
  const int b  = blockIdx.z;
  const int n0 = blockIdx.x * 128;
  const int m0 = blockIdx.y * 128 + wave * 16;

  const float* Xb = Xin  + (size_t)b * K * N;
  float*       Yb = Yout + (size_t)b * O * N;

  // C initialized with bias (channel m0 + r + 8*half, identical for every N)
  v8f acc[8];
#pragma unroll
  for (int r = 0; r < 8; ++r) {
    const float bv = bias[m0 + r + 8 * half];
#pragma unroll
    for (int t = 0; t < 8; ++t) acc[t][r] = bv;
  }

  // per-thread staging registers (16 rows x 128 cols / 256 threads = 8)
  float ldreg[8];
  const int lrow = tid >> 4;          // unused; loop rows are flat>>7, flat = tid + i*256
  const int lcolb = tid & 127;        // base col component
#pragma unroll
  for (int i = 0; i < 8; ++i) {
    const int flat = tid + i * 256;
    const int r = flat >> 7, col = flat & 127;
    ldreg[i] = Xb[(size_t)r * N + (n0 + col)];
  }
  (void)lrow; (void)lcolb;
#pragma unroll
  for (int i = 0; i < 8; ++i) {
    const int flat = tid + i * 256;
    const int r = flat >> 7, col = flat & 127;
    sB[0][(((r >> 1) * 128) + col) * 2 + (r & 1)] = ldreg[i];
  }
  __syncthreads();

  const int nchunks = K >> 4;
  for (int ci = 0; ci < nchunks; ++ci) {
    const int k0 = ci << 4;
    const bool more = (ci + 1) < nchunks;

    // prefetch next chunk into registers (overlaps with WMMA below)
    if (more) {
#pragma unroll
      for (int i = 0; i < 8; ++i) {
        const int flat = tid + i * 256;
        const int r = flat >> 7, col = flat & 127;
        ldreg[i] = Xb[(size_t)(k0 + 16 + r) * N + (n0 + col)];
      }
    }

    // hoist all 4 A fragments for this chunk (issued together)
    v2f afrag[4];
#pragma unroll
    for (int kk4 = 0; kk4 < 4; ++kk4) {
      const float* wr = Wm + (size_t)(m0 + lid) * K + (k0 + kk4 * 4 + 2 * half);
      afrag[kk4].x = wr[0];
      afrag[kk4].y = wr[1];
    }

    const float* sbuf = sB[ci & 1];
#pragma unroll
    for (int kk4 = 0; kk4 < 4; ++kk4) {
      const int p = kk4 * 2 + half;   // K-pair index = (kk + 2*half)/2
#pragma unroll
      for (int t = 0; t < 8; ++t) {
        const v2f bv = *(const v2f*)&sbuf[(p * 128 + t * 16 + lid) * 2];
        acc[t] = __builtin_amdgcn_wmma_f32_16x16x4_f32(
            false, afrag[kk4], false, bv, (short)0, acc[t], false, false);
      }
    }

    if (more) {
#pragma unroll
      for (int i = 0; i < 8; ++i) {
        const int flat = tid + i * 256;
        const int r = flat >> 7, col = flat & 127;
        sB[(ci + 1) & 1][(((r >> 1) * 128) + col) * 2 + (r & 1)] = ldreg[i];
      }
      __syncthreads();
    }
  }

  // D layout: VGPR r -> M = m0 + r + 8*half, N = n0 + t*16 + lid
#pragma unroll
  for (int t = 0; t < 8; ++t)
#pragma unroll
    for (int r = 0; r < 8; ++r)
      Yb[(size_t)(m0 + r + 8 * half) * N + (n0 + t * 16 + lid)] = acc[t][r];
}

// ---------------------------------------------------------------------
// Kernel 4: per-channel sum / sumsq over (B, N), deterministic LDS tree.
// One block per channel. N is always N1_ (8192) -> shifts, no divides.
// ---------------------------------------------------------------------
__global__ __launch_bounds__(256) void bn_stats_kernel(
    const float* __restrict__ Y, float* __restrict__ stats, int C) {
  __shared__ float ss[256];
  __shared__ float sq[256];
  const int c = blockIdx.x;
  float s = 0.0f, q = 0.0f;
  for (int i = threadIdx.x; i < B_ * N1_; i += 256) {
    const int bb = i >> 13;          // / 8192
    const int n  = i & (N1_ - 1);
    const float v = Y[((size_t)bb * C + c) * N1_ + n];
    s += v;
    q += v * v;
  }
  ss[threadIdx.x] = s;
  sq[threadIdx.x] = q;
  __syncthreads();
  for (int st = 128; st > 0; st >>= 1) {
    if (threadIdx.x < st) {
      ss[threadIdx.x] += ss[threadIdx.x + st];
      sq[threadIdx.x] += sq[threadIdx.x + st];
    }
    __syncthreads();
  }
  if (threadIdx.x == 0) {
    stats[c]     = ss[0];
    stats[C + c] = sq[0];
  }
}

// ---------------------------------------------------------------------
// Kernel 5: apply BN (biased batch stats) + ReLU, in place.
// One block per (channel, batch).
// ---------------------------------------------------------------------
__global__ __launch_bounds__(256) void bn_apply_kernel(
    float* __restrict__ Y, const float* __restrict__ stats,
    const float* __restrict__ g, const float* __restrict__ be, int C) {
  const int c = blockIdx.x;
  const int b = blockIdx.y;
  const float invn = 1.0f / (float)(B_ * N1_);
  const float mean = stats[c] * invn;
  const float var  = stats[C + c] * invn - mean * mean;
  const float sc   = g[c] * rsqrtf(var + EPS_BN);
  const float sh   = be[c] - mean * sc;
  float* row = Y + ((size_t)b * C + c) * N1_;
  for (int n = threadIdx.x; n < N1_; n += 256) {
    const float v = row[n] * sc + sh;
    row[n] = v > 0.0f ? v : 0.0f;
  }
}

// ---------------------------------------------------------------------
extern "C" void kernel_launch(void* const* d_in, const int* in_sizes, int n_in,
                              void* d_out, int out_size, void* d_ws, size_t ws_size,
                              hipStream_t stream) {
  const float* xyz1    = (const float*)d_in[0];
  const float* xyz2    = (const float*)d_in[1];
  const float* points1 = (const float*)d_in[2];
  const float* points2 = (const float*)d_in[3];
  const float* w0      = (const float*)d_in[4];
  const float* b0      = (const float*)d_in[5];
  const float* g0      = (const float*)d_in[6];
  const float* be0     = (const float*)d_in[7];
  const float* w1      = (const float*)d_in[8];
  const float* b1      = (const float*)d_in[9];
  const float* g1      = (const float*)d_in[10];
  const float* be1     = (const float*)d_in[11];
  float* out = (float*)d_out;

  // workspace layout (bytes)
  char* ws = (char*)d_ws;
  const size_t idx_bytes = (size_t)B_ * N1_ * 3 * sizeof(int);
  const size_t w_bytes   = (size_t)B_ * N1_ * 3 * sizeof(float);
  const size_t X_bytes   = (size_t)B_ * CX_ * N1_ * sizeof(float);
  const size_t Y0_bytes  = (size_t)B_ * O0_ * N1_ * sizeof(float);
  int*   idx   = (int*)ws;
  float* wgt   = (float*)(ws + idx_bytes);
  float* X     = (float*)(ws + idx_bytes + w_bytes);
  float* Y0    = (float*)(ws + idx_bytes + w_bytes + X_bytes);
  float* stats = (float*)(ws + idx_bytes + w_bytes + X_bytes + Y0_bytes);
  (void)ws_size; (void)in_sizes; (void)n_in; (void)out_size;

  // 1) three_nn + weights
  knn3_kernel<<<dim3(N1_ / 256, B_), 256, 0, stream>>>(xyz1, xyz2, idx, wgt);

  // 2) X = concat(points1, interp(points2))
  buildx_kernel<<<dim3(CX_, B_), 256, 0, stream>>>(points1, points2, idx, wgt, X);

  // 3) layer 0 GEMM (WMMA f32) + bias
  gemm_bias_kernel<<<dim3(N1_ / 128, O0_ / 128, B_), 256, 0, stream>>>(
      w0, b0, X, Y0, O0_, CX_, N1_);

  // 4/5) BN0 + ReLU (deterministic two-pass)
  bn_stats_kernel<<<O0_, 256, 0, stream>>>(Y0, stats, O0_);
  bn_apply_kernel<<<dim3(O0_, B_), 256, 0, stream>>>(Y0, stats, g0, be0, O0_);

  // 6) layer 1 GEMM (WMMA f32) + bias -> d_out (pre-BN)
  gemm_bias_kernel<<<dim3(N1_ / 128, O1_ / 128, B_), 256, 0, stream>>>(
      w1, b1, Y0, out, O1_, O0_, N1_);

  // 7/8) BN1 + ReLU in place on d_out
  bn_stats_kernel<<<O1_, 256, 0, stream>>>(out, stats, O1_);
  bn_apply_kernel<<<dim3(O1_, B_), 256, 0, stream>>>(out, stats, g1, be1, O1_);
}